// DQN_31258771980824
// MI455X (gfx1250) — compile-verified
//
#include <hip/hip_runtime.h>
#include <hip/hip_bf16.h>

typedef __attribute__((ext_vector_type(2))) float v2f;
typedef __attribute__((ext_vector_type(8))) float v8f;

#define N_FEAT 128

// ---- degree / normalization -------------------------------------------------

__global__ void deg_init_kernel(float* __restrict__ deg, int n) {
  int i = blockIdx.x * blockDim.x + threadIdx.x;
  if (i < n) deg[i] = 1.0f;  // self-loop contributes 1 to every in-degree
}

__global__ void deg_edges_kernel(const int* __restrict__ col,
                                 float* __restrict__ deg, int E) {
  int e = blockIdx.x * blockDim.x + threadIdx.x;
  if (e < E) atomicAdd(&deg[col[e]], 1.0f);
}

__global__ void dinv_kernel(float* __restrict__ deg, int n) {
  int i = blockIdx.x * blockDim.x + threadIdx.x;
  if (i < n) {
    float d = deg[i];
    deg[i] = d > 0.0f ? rsqrtf(d) : 0.0f;  // in place: deg -> dinv
  }
}

// ---- dense GEMM via f32 WMMA: C[N,128] = A[N,128] @ W[128,128] --------------
// One wave per 16x16 output tile; K swept in steps of 4 with
// V_WMMA_F32_16X16X4_F32 (exact f32, matches reference numerics).
__global__ void __launch_bounds__(256) gemm_wmma_f32_kernel(
    const float* __restrict__ A, const float* __restrict__ W,
    float* __restrict__ C, int nRowTiles) {
  const int wave = threadIdx.x >> 5;
  const int lane = threadIdx.x & 31;
  const int mt = blockIdx.x * 8 + wave;  // row tile (wave-uniform)
  const int nt = blockIdx.y;             // col tile, 0..7
  if (mt >= nRowTiles) return;           // uniform per wave -> EXEC all 0 or all 1
  const int half = lane >> 4;            // 0: K=k,k+1 ; 1: K=k+2,k+3 halves
  const int l15 = lane & 15;

  // A fragment: lane (half,l15) holds row mt*16+l15, cols kk,kk+1
  const float* arow = A + (size_t)(mt * 16 + l15) * N_FEAT;
  // B fragment: lane (half,l15) holds col nt*16+l15, rows kk,kk+1
  const float* wcol = W + nt * 16 + l15;

  v8f acc = {};
#pragma unroll 8
  for (int k = 0; k < N_FEAT; k += 4) {
    const int kk = k + 2 * half;
    v2f a, b;
    a[0] = arow[kk];
    a[1] = arow[kk + 1];
    b[0] = wcol[(size_t)kk * N_FEAT];
    b[1] = wcol[(size_t)(kk + 1) * N_FEAT];
    acc = __builtin_amdgcn_wmma_f32_16x16x4_f32(
        /*neg_a=*/false, a, /*neg_b=*/false, b,
        /*c_mod=*/(short)0, acc, /*reuse_a=*/false, /*reuse_b=*/false);
  }

  // C/D layout: VGPR r -> M = r + 8*half, N = l15
  const int mbase = mt * 16 + half * 8;
#pragma unroll
  for (int r = 0; r < 8; ++r)
    C[(size_t)(mbase + r) * N_FEAT + nt * 16 + l15] = acc[r];
}

// ---- aggregation ------------------------------------------------------------

// agg[i,:] = dinv[i]^2 * xw[i,:]   (self-loop term; also zero-initializes agg)
__global__ void init_agg_kernel(const float* __restrict__ xw,
                                const float* __restrict__ dinv,
                                float* __restrict__ agg, int n) {
  int idx = blockIdx.x * blockDim.x + threadIdx.x;  // float4 index
  int total = n * (N_FEAT / 4);
  if (idx >= total) return;
  int i = idx >> 5;  // row = idx / (128/4)
  float d = dinv[i];
  float s = d * d;
  float4 v = ((const float4*)xw)[idx];
  v.x *= s; v.y *= s; v.z *= s; v.w *= s;
  ((float4*)agg)[idx] = v;
}

// one wave per edge: float4 gather of xw[row,:], scaled atomic-add into agg[col,:]
__global__ void __launch_bounds__(256) scatter_kernel(
    const int* __restrict__ row, const int* __restrict__ colv,
    const float* __restrict__ dinv, const float* __restrict__ xw,
    float* __restrict__ agg, int E) {
  const int wid = blockIdx.x * 8 + (threadIdx.x >> 5);
  if (wid >= E) return;
  const int lane = threadIdx.x & 31;
  const int r = row[wid];
  const int c = colv[wid];
  const float nrm = dinv[r] * dinv[c];
  const float4 v = ((const float4*)(xw + (size_t)r * N_FEAT))[lane];
  float* dst = agg + (size_t)c * N_FEAT + lane * 4;
  atomicAdd(dst + 0, nrm * v.x);
  atomicAdd(dst + 1, nrm * v.y);
  atomicAdd(dst + 2, nrm * v.z);
  atomicAdd(dst + 3, nrm * v.w);
}

__global__ void bias_relu_kernel(float* __restrict__ h,
                                 const float* __restrict__ b, int n) {
  int idx = blockIdx.x * blockDim.x + threadIdx.x;  // float4 index
  int total = n * (N_FEAT / 4);
  if (idx >= total) return;
  int f4 = idx & 31;
  float4 bb = ((const float4*)b)[f4];
  float4 v = ((float4*)h)[idx];
  v.x = fmaxf(v.x + bb.x, 0.0f);
  v.y = fmaxf(v.y + bb.y, 0.0f);
  v.z = fmaxf(v.z + bb.z, 0.0f);
  v.w = fmaxf(v.w + bb.w, 0.0f);
  ((float4*)h)[idx] = v;
}

// ---- pooling + readout ------------------------------------------------------

__global__ void zero_kernel(float* __restrict__ p, int n) {
  int i = blockIdx.x * blockDim.x + threadIdx.x;
  if (i < n) p[i] = 0.0f;
}

__global__ void colsum_kernel(const float* __restrict__ h,
                              float* __restrict__ g, int n) {
  const int f = threadIdx.x;  // blockDim.x == 128
  float s = 0.0f;
  for (int i = blockIdx.x; i < n; i += gridDim.x)
    s += h[(size_t)i * N_FEAT + f];
  atomicAdd(&g[f], s);
}

// out[a] = (sum_f g[f]/N * Wl[f,a]) + bl[a],  A = 16
__global__ void readout_kernel(const float* __restrict__ g,
                               const float* __restrict__ Wl,
                               const float* __restrict__ bl,
                               float* __restrict__ out, float invn) {
  int a = threadIdx.x;
  if (a < 16) {
    float s = bl[a];
    for (int f = 0; f < N_FEAT; ++f) s += g[f] * invn * Wl[f * 16 + a];
    out[a] = s;
  }
}

// ---- driver -----------------------------------------------------------------

extern "C" void kernel_launch(void* const* d_in, const int* in_sizes, int n_in,
                              void* d_out, int out_size, void* d_ws,
                              size_t ws_size, hipStream_t stream) {
  const float* x  = (const float*)d_in[0];
  const int*   ei = (const int*)d_in[1];   // [2, E] flat: rows then cols
  const float* W1 = (const float*)d_in[2];
  const float* b1 = (const float*)d_in[3];
  const float* W2 = (const float*)d_in[4];
  const float* b2 = (const float*)d_in[5];
  const float* Wl = (const float*)d_in[6];
  const float* bl = (const float*)d_in[7];
  float* out = (float*)d_out;

  const int N = in_sizes[0] / N_FEAT;  // 100000 (multiple of 16)
  const int E = in_sizes[1] / 2;       // 1600000
  const int* row = ei;
  const int* col = ei + E;

  // workspace: bufA | bufB | dinv | g
  float* bufA = (float*)d_ws;
  float* bufB = bufA + (size_t)N * N_FEAT;
  float* dinv = bufB + (size_t)N * N_FEAT;
  float* g    = dinv + N;

  const int nRowTiles = N / 16;
  dim3 gemmGrid((nRowTiles + 7) / 8, N_FEAT / 16);
  const int elem4 = N * (N_FEAT / 4);

  // normalization coefficients
  deg_init_kernel<<<(N + 255) / 256, 256, 0, stream>>>(dinv, N);
  deg_edges_kernel<<<(E + 255) / 256, 256, 0, stream>>>(col, dinv, E);
  dinv_kernel<<<(N + 255) / 256, 256, 0, stream>>>(dinv, N);

  // layer 1
  gemm_wmma_f32_kernel<<<gemmGrid, 256, 0, stream>>>(x, W1, bufA, nRowTiles);
  init_agg_kernel<<<(elem4 + 255) / 256, 256, 0, stream>>>(bufA, dinv, bufB, N);
  scatter_kernel<<<(E + 7) / 8, 256, 0, stream>>>(row, col, dinv, bufA, bufB, E);
  bias_relu_kernel<<<(elem4 + 255) / 256, 256, 0, stream>>>(bufB, b1, N);

  // layer 2
  gemm_wmma_f32_kernel<<<gemmGrid, 256, 0, stream>>>(bufB, W2, bufA, nRowTiles);
  init_agg_kernel<<<(elem4 + 255) / 256, 256, 0, stream>>>(bufA, dinv, bufB, N);
  scatter_kernel<<<(E + 7) / 8, 256, 0, stream>>>(row, col, dinv, bufA, bufB, E);
  bias_relu_kernel<<<(elem4 + 255) / 256, 256, 0, stream>>>(bufB, b2, N);

  // mean pool + linear readout
  zero_kernel<<<1, N_FEAT, 0, stream>>>(g, N_FEAT);
  colsum_kernel<<<512, N_FEAT, 0, stream>>>(bufB, g, N);
  readout_kernel<<<1, 32, 0, stream>>>(g, Wl, bl, out, 1.0f / (float)N);
}